// Mamba2Block_38697655337233
// MI455X (gfx1250) — compile-verified
//
#include <hip/hip_runtime.h>
#include <math.h>
#include <stdint.h>

// ---------------- Model constants ----------------
#define D_MODEL   512
#define D_STATE   64
#define D_CONV    4
#define NHEADS    16
#define HEADDIM   64
#define D_INNER   1024
#define E_PROJ    2192           // 2*D_INNER + 2*D_STATE + NHEADS
#define BATCH     2
#define SEQLEN    2048
#define M_TOT     (BATCH*SEQLEN) // 4096
#define OFF_Z     0
#define OFF_XS    1024
#define OFF_B     2048
#define OFF_C     2112
#define OFF_DT    2176
#define LC        16             // chunk length
#define NCHUNK    (SEQLEN/LC)    // 128

typedef __attribute__((ext_vector_type(2))) float v2f;
typedef __attribute__((ext_vector_type(8))) float v8f;
typedef __attribute__((ext_vector_type(4))) int   v4i;

__device__ __forceinline__ v8f wmma_f32_k4(v2f a, v2f b, v8f c) {
  // D = A(16x4) * B(4x16) + C, fp32, wave32
  return __builtin_amdgcn_wmma_f32_16x16x4_f32(false, a, false, b, (short)0, c,
                                               false, false);
}

// ---------------- CDNA5 async global->LDS copies (ASYNCcnt) ----------------
#if defined(__gfx1250__) &&                                              \
    __has_builtin(__builtin_amdgcn_global_load_async_to_lds_b128) &&     \
    __has_builtin(__builtin_amdgcn_global_load_async_to_lds_b32)
#define USE_ASYNC_LDS 1
#else
#define USE_ASYNC_LDS 0
#endif

// AS1 (= HIP "__device__") global pointers; AS3 LDS pointers.
typedef __attribute__((address_space(1))) v4i* gp_v4i;
typedef __attribute__((address_space(3))) v4i* lp_v4i;
typedef __attribute__((address_space(1))) int* gp_i32;
typedef __attribute__((address_space(3))) int* lp_i32;

__device__ __forceinline__ void async_b128(const void* g, void* l) {
#if USE_ASYNC_LDS
  // AS1 pointer value == generic global address; AS3 pointer == low 32 bits
  // of the generic LDS-aperture address.
  __builtin_amdgcn_global_load_async_to_lds_b128(
      (gp_v4i)(uintptr_t)g, (lp_v4i)(uint32_t)(uintptr_t)l, 0, 0);
#else
  *(float4*)l = *(const float4*)g;
#endif
}

__device__ __forceinline__ void async_b32(const void* g, void* l) {
#if USE_ASYNC_LDS
  __builtin_amdgcn_global_load_async_to_lds_b32(
      (gp_i32)(uintptr_t)g, (lp_i32)(uint32_t)(uintptr_t)l, 0, 0);
#else
  *(float*)l = *(const float*)g;
#endif
}

__device__ __forceinline__ void async_wait_all() {
#if USE_ASYNC_LDS
#if __has_builtin(__builtin_amdgcn_s_wait_asynccnt)
  __builtin_amdgcn_s_wait_asynccnt(0);
#else
  asm volatile("s_wait_asynccnt 0" ::: "memory");
#endif
#endif
}

// =====================================================================
// Generic NT GEMM:  C[m,n] = sum_k A[m,k] * B[n,k]
// One wave per 16x16 output tile; fp32 WMMA (16x16x4).
// =====================================================================
__global__ __launch_bounds__(128)
void gemm_nt_wmma(const float* __restrict__ A, const float* __restrict__ B,
                  float* __restrict__ C, int K, int lda, int ldb, int ldc,
                  int ntiles) {
  const int lane = threadIdx.x & 31;
  const int wave = threadIdx.x >> 5;
  const int mt   = blockIdx.x;
  const int nt   = blockIdx.y * 4 + wave;
  if (nt >= ntiles) return;                 // wave-uniform exit

  const int l15  = lane & 15;
  const int hi16 = lane >> 4;               // 0 or 1
  const int koff = hi16 << 1;               // 0 or 2

  const float* arow = A + (size_t)(mt * 16 + l15) * lda;
  const float* brow = B + (size_t)(nt * 16 + l15) * ldb;

  v8f acc = {0.f, 0.f, 0.f, 0.f, 0.f, 0.f, 0.f, 0.f};
  for (int k0 = 0; k0 < K; k0 += 4) {
    if ((k0 & 63) == 0 && k0 + 64 < K) {
      __builtin_prefetch(arow + k0 + 64, 0, 1);   // global_prefetch_b8
      __builtin_prefetch(brow + k0 + 64, 0, 1);
    }
    v2f a, b;
    a.x = arow[k0 + koff];     a.y = arow[k0 + koff + 1];
    b.x = brow[k0 + koff];     b.y = brow[k0 + koff + 1];
    acc = wmma_f32_k4(a, b, acc);
  }
  const int col = nt * 16 + l15;
#pragma unroll
  for (int i = 0; i < 8; ++i) {
    const int row = mt * 16 + i + hi16 * 8;
    C[(size_t)row * ldc + col] = acc[i];
  }
}

// =====================================================================
// Depthwise causal conv1d (width 4) + bias + SiLU over the xs slice of proj.
// =====================================================================
__global__ void conv_silu_kernel(const float* __restrict__ proj,
                                 const float* __restrict__ conv_w,
                                 const float* __restrict__ conv_b,
                                 float* __restrict__ xconv) {
  const int idx = blockIdx.x * blockDim.x + threadIdx.x; // < M_TOT*D_INNER
  const int c = idx & (D_INNER - 1);
  const int g = idx >> 10;
  const int t = g & (SEQLEN - 1);
  const int b = g >> 11;
  float acc = conv_b[c];
#pragma unroll
  for (int j = 0; j < D_CONV; ++j) {
    const int tt = t - (D_CONV - 1) + j;
    if (tt >= 0)
      acc += conv_w[c * D_CONV + j] *
             proj[(size_t)(b * SEQLEN + tt) * E_PROJ + OFF_XS + c];
  }
  const float s = acc / (1.f + expf(-acc)); // silu
  xconv[(size_t)g * D_INNER + c] = s;
}

// =====================================================================
// dt -> dA = exp(softplus(dt + dt_bias) * (-exp(A_log)))
// =====================================================================
__global__ void da_kernel(const float* __restrict__ proj,
                          const float* __restrict__ A_log,
                          const float* __restrict__ dt_bias,
                          float* __restrict__ dAbuf) {
  const int idx = blockIdx.x * blockDim.x + threadIdx.x; // < M_TOT*NHEADS
  const int h = idx & (NHEADS - 1);
  const int g = idx >> 4;
  const float dtv = proj[(size_t)g * E_PROJ + OFF_DT + h] + dt_bias[h];
  const float sp = (dtv > 20.f) ? dtv : log1pf(expf(dtv));
  const float A = -expf(A_log[h]);
  dAbuf[idx] = expf(sp * A);
}

// =====================================================================
// Chunked selective scan, one block per (batch, head), 4 waves / 128 threads.
// State S (64x64 fp32) in LDS; chunk tiles double-buffered with async
// global->LDS copies so chunk ch+1's DMA overlaps chunk ch's WMMA compute.
// =====================================================================
__global__ __launch_bounds__(128)
void scan_chunked_kernel(const float* __restrict__ proj,
                         const float* __restrict__ xconv,
                         const float* __restrict__ dAbuf,
                         float* __restrict__ ybuf) {
  __shared__ __align__(16) float sB[2][LC * 64];
  __shared__ __align__(16) float sC[2][LC * 64];
  __shared__ __align__(16) float sX[2][LC * 64];
  __shared__ __align__(16) float sS[64 * 64];
  __shared__ float sM[LC * LC];
  __shared__ float sCum[LC];
  __shared__ float sRat[LC];
  __shared__ __align__(16) float sAh[2][LC];
  __shared__ float sCumEnd;

  const int tid   = threadIdx.x;
  const int lane  = tid & 31;
  const int wave  = tid >> 5;
  const int bh    = blockIdx.x;
  const int batch = bh >> 4;
  const int h     = bh & 15;

  const int l15     = lane & 15;
  const int hi16    = lane >> 4;
  const int kget    = hi16 << 1;      // 0 or 2
  const int colbase = wave * 16;      // p-strip per wave

  for (int i = tid; i < 64 * 64; i += 128) sS[i] = 0.f;

  // issue async loads for one chunk's tiles into buffer `buf`
  const auto issue_chunk = [&](int ch2, int buf) {
#pragma unroll
    for (int r = 0; r < 2; ++r) {
      const int q  = tid + r * 128;          // 16B-chunk id, 0..255
      const int t  = q >> 4;
      const int n4 = (q & 15) << 2;
      const size_t tok = (size_t)batch * SEQLEN + (size_t)ch2 * LC + t;
      const size_t row = tok * E_PROJ;
      async_b128(&proj[row + OFF_B + n4], &sB[buf][q << 2]);
      async_b128(&proj[row + OFF_C + n4], &sC[buf][q << 2]);
      async_b128(&xconv[tok * D_INNER + h * HEADDIM + n4], &sX[buf][q << 2]);
    }
    if (tid < LC) {
      const size_t tok = (size_t)batch * SEQLEN + (size_t)ch2 * LC + tid;
      async_b32(&dAbuf[tok * NHEADS + h], &sAh[buf][tid]);
    }
  };

  issue_chunk(0, 0);

  for (int ch = 0; ch < NCHUNK; ++ch) {
    const int buf = ch & 1;
    const size_t tok0 = (size_t)batch * SEQLEN + (size_t)ch * LC;

    async_wait_all();     // this wave's pending async LDS writes done
    __syncthreads();      // tiles for chunk `ch` visible to all waves

    // overlap the next chunk's DMA with this chunk's compute
    if (ch + 1 < NCHUNK) issue_chunk(ch + 1, buf ^ 1);

    if (tid == 0) {
      float c = 1.f;
      for (int t = 0; t < LC; ++t) { c *= sAh[buf][t]; sCum[t] = c; }
      sCumEnd = c;
      for (int t = 0; t < LC; ++t) sRat[t] = c / sCum[t];
    }
    __syncthreads();

    // ---- G = C @ B^T (wave 0), build masked decay matrix ----
    if (wave == 0) {
      v8f g = {0.f, 0.f, 0.f, 0.f, 0.f, 0.f, 0.f, 0.f};
#pragma unroll
      for (int k0 = 0; k0 < 64; k0 += 4) {
        v2f a, b;
        a.x = sC[buf][l15 * 64 + k0 + kget];
        a.y = sC[buf][l15 * 64 + k0 + kget + 1];
        b.x = sB[buf][l15 * 64 + k0 + kget];
        b.y = sB[buf][l15 * 64 + k0 + kget + 1];
        g = wmma_f32_k4(a, b, g);
      }
#pragma unroll
      for (int i = 0; i < 8; ++i) {
        const int t = i + hi16 * 8;   // row (time)
        const int s = l15;            // col (source time)
        sM[t * LC + s] = (s <= t) ? g[i] * (sCum[t] / sCum[s]) : 0.f;
      }
    }
    __syncthreads();

    // ---- Y = diag(cum)*(C@S) + M@X ; gated write ----
    {
      v8f acc = {0.f, 0.f, 0.f, 0.f, 0.f, 0.f, 0.f, 0.f};
#pragma unroll
      for (int k0 = 0; k0 < 64; k0 += 4) {
        v2f a, b;
        a.x = sC[buf][l15 * 64 + k0 + kget];
        a.y = sC[buf][l15 * 64 + k0 + kget + 1];
        b.x = sS[(k0 + kget) * 64 + colbase + l15];
        b.y = sS[(k0 + kget + 1) * 64 + colbase + l15];
        acc = wmma_f32_k4(a, b, acc);
      }
#pragma unroll
      for (int i = 0; i < 8; ++i) acc[i] *= sCum[i + hi16 * 8];
#pragma unroll
      for (int k0 = 0; k0 < LC; k0 += 4) {
        v2f a, b;
        a.x = sM[l15 * LC + k0 + kget];
        a.y = sM[l15 * LC + k0 + kget + 1];
        b.x = sX[buf][(k0 + kget) * 64 + colbase + l15];
        b.y = sX[buf][(k0 + kget + 1) * 64 + colbase + l15];
        acc = wmma_f32_k4(a, b, acc);
      }
      const int p = colbase + l15;
#pragma unroll
      for (int i = 0; i < 8; ++i) {
        const int t = i + hi16 * 8;
        const size_t g = tok0 + t;
        const float z = proj[g * E_PROJ + OFF_Z + h * HEADDIM + p];
        const float sil = z / (1.f + expf(-z));
        ybuf[g * D_INNER + h * HEADDIM + p] = acc[i] * sil;
      }
    }
    __syncthreads();

    // ---- S = cumEnd*S + B'^T @ X  (per-wave p-strip) ----
    {
      const float ce = sCumEnd;
#pragma unroll
      for (int mt2 = 0; mt2 < 4; ++mt2) {
        v8f acc;
#pragma unroll
        for (int i = 0; i < 8; ++i)
          acc[i] = ce * sS[(mt2 * 16 + i + hi16 * 8) * 64 + colbase + l15];
#pragma unroll
        for (int k0 = 0; k0 < LC; k0 += 4) {
          const int s0 = k0 + kget;
          v2f a, b;
          a.x = sB[buf][s0 * 64 + mt2 * 16 + l15] * sRat[s0];
          a.y = sB[buf][(s0 + 1) * 64 + mt2 * 16 + l15] * sRat[s0 + 1];
          b.x = sX[buf][s0 * 64 + colbase + l15];
          b.y = sX[buf][(s0 + 1) * 64 + colbase + l15];
          acc = wmma_f32_k4(a, b, acc);
        }
#pragma unroll
        for (int i = 0; i < 8; ++i)
          sS[(mt2 * 16 + i + hi16 * 8) * 64 + colbase + l15] = acc[i];
      }
    }
    __syncthreads();
  }
}

// =====================================================================
extern "C" void kernel_launch(void* const* d_in, const int* in_sizes, int n_in,
                              void* d_out, int out_size, void* d_ws,
                              size_t ws_size, hipStream_t stream) {
  (void)in_sizes; (void)n_in; (void)out_size; (void)ws_size;
  const float* x       = (const float*)d_in[0];
  const float* W_in    = (const float*)d_in[1];
  const float* conv_w  = (const float*)d_in[2];
  const float* conv_b  = (const float*)d_in[3];
  const float* A_log   = (const float*)d_in[4];
  const float* dt_bias = (const float*)d_in[5];
  const float* W_out   = (const float*)d_in[6];
  float* out = (float*)d_out;

  // workspace carve-up (fp32)
  float* proj  = (float*)d_ws;                         // M_TOT x E_PROJ
  float* xconv = proj  + (size_t)M_TOT * E_PROJ;       // M_TOT x D_INNER
  float* dAbuf = xconv + (size_t)M_TOT * D_INNER;      // M_TOT x NHEADS
  float* ybuf  = dAbuf + (size_t)M_TOT * NHEADS;       // M_TOT x D_INNER

  // 1) in_proj: proj = x @ W_in^T   (M=4096, N=2192, K=512)
  {
    const int ntiles = E_PROJ / 16;                    // 137
    dim3 grid(M_TOT / 16, (ntiles + 3) / 4);
    gemm_nt_wmma<<<grid, 128, 0, stream>>>(x, W_in, proj, D_MODEL, D_MODEL,
                                           D_MODEL, E_PROJ, ntiles);
  }
  // 2) depthwise conv + SiLU
  {
    const int n = M_TOT * D_INNER;
    conv_silu_kernel<<<n / 256, 256, 0, stream>>>(proj, conv_w, conv_b, xconv);
  }
  // 2b) dt -> dA
  {
    const int n = M_TOT * NHEADS;
    da_kernel<<<n / 256, 256, 0, stream>>>(proj, A_log, dt_bias, dAbuf);
  }
  // 3) chunked selective scan + gating (async double-buffered tiles)
  scan_chunked_kernel<<<BATCH * NHEADS, 128, 0, stream>>>(proj, xconv, dAbuf,
                                                          ybuf);
  // 4) out_proj: out = ybuf @ W_out^T  (M=4096, N=512, K=1024)
  {
    const int ntiles = D_MODEL / 16;                   // 32
    dim3 grid(M_TOT / 16, ntiles / 4);
    gemm_nt_wmma<<<grid, 128, 0, stream>>>(ybuf, W_out, out, D_INNER, D_INNER,
                                           D_INNER, D_MODEL, ntiles);
  }
}